// MultiScaleAttention_51943334478247
// MI455X (gfx1250) — compile-verified
//
#include <hip/hip_runtime.h>
#include <math.h>

#define B_   16
#define C_   64
#define H_   256
#define W_   256
#define HW_  (H_ * W_)
#define HID_ 32

typedef __attribute__((ext_vector_type(2))) float v2f;
typedef __attribute__((ext_vector_type(8))) float v8f;

__device__ __forceinline__ float sigmoidf_(float v) {
    return 1.0f / (1.0f + expf(-v));
}

// ---------------------------------------------------------------------------
// k0: zero the per-batch pooled accumulators (graph-safe init each launch)
// ---------------------------------------------------------------------------
__global__ void k0_init(float* __restrict__ pooledAcc) {
    if (threadIdx.x < B_) pooledAcc[threadIdx.x] = 0.0f;
}

// ---------------------------------------------------------------------------
// k1: single streaming pass over x producing per-pixel
//     mean_c(x), max_c(x), dot[p] = sum_c wr[c]*x[c,p]
// float4-vectorized over 4 consecutive pixels; coalesced per channel plane.
// ---------------------------------------------------------------------------
__global__ void k1_stats(const float* __restrict__ x, const float* __restrict__ wr,
                         float* __restrict__ meanp, float* __restrict__ maxp,
                         float* __restrict__ dotp) {
    __shared__ float swr[C_];
    int t = threadIdx.x;
    if (t < C_) swr[t] = wr[t];
    __syncthreads();

    long q  = (long)blockIdx.x * blockDim.x + t;   // quad (float4) pixel index
    long p  = q * 4;                                // pixel index in [0, B*HW)
    int  b  = (int)(p / HW_);
    int  hw = (int)(p % HW_);
    const float* xb = x + (long)b * C_ * HW_ + hw;

    float4 s = make_float4(0.f, 0.f, 0.f, 0.f);
    float4 m = make_float4(-INFINITY, -INFINITY, -INFINITY, -INFINITY);
    float4 d = make_float4(0.f, 0.f, 0.f, 0.f);

    for (int c = 0; c < C_; ++c) {
        if (c < C_ - 8) __builtin_prefetch(xb + (long)(c + 8) * HW_, 0, 1);
        float4 v = *(const float4*)(xb + (long)c * HW_);
        float  w = swr[c];
        s.x += v.x; s.y += v.y; s.z += v.z; s.w += v.w;
        m.x = fmaxf(m.x, v.x); m.y = fmaxf(m.y, v.y);
        m.z = fmaxf(m.z, v.z); m.w = fmaxf(m.w, v.w);
        d.x += w * v.x; d.y += w * v.y; d.z += w * v.z; d.w += w * v.w;
    }
    const float inv = 1.0f / (float)C_;
    s.x *= inv; s.y *= inv; s.z *= inv; s.w *= inv;
    ((float4*)meanp)[q] = s;
    ((float4*)maxp)[q]  = m;
    ((float4*)dotp)[q]  = d;
}

// ---------------------------------------------------------------------------
// k2: spatial attention. 16x16 output tile per block, LDS halo tile (20x20x2ch),
//     3x3 and 5x5 convs + sigmoid + 1x1 fuse -> scale in [0.2, 1.0].
//     Also accumulates pooled[b] += sum_p scale[p] * dot[p] (block reduce +
//     one global_atomic_add_f32 per block).
// ---------------------------------------------------------------------------
__global__ void k2_conv(const float* __restrict__ meanp, const float* __restrict__ maxp,
                        const float* __restrict__ dotp,
                        const float* __restrict__ w3, const float* __restrict__ b3,
                        const float* __restrict__ w5, const float* __restrict__ b5,
                        const float* __restrict__ wf, const float* __restrict__ bf,
                        float* __restrict__ fused, float* __restrict__ pooledAcc) {
    __shared__ float tmn[20][21];
    __shared__ float tmx[20][21];
    __shared__ float wgt[73];   // [0,18)=w3  [18,68)=w5  [68,70)=wf  70=b3 71=b5 72=bf
    __shared__ float red[256];

    int lx = threadIdx.x, ly = threadIdx.y;
    int tid = ly * 16 + lx;
    int b  = blockIdx.z;
    int bx = blockIdx.x * 16, by = blockIdx.y * 16;

    if      (tid < 18) wgt[tid] = w3[tid];
    else if (tid < 68) wgt[tid] = w5[tid - 18];
    else if (tid < 70) wgt[tid] = wf[tid - 68];
    else if (tid == 70) wgt[70] = b3[0];
    else if (tid == 71) wgt[71] = b5[0];
    else if (tid == 72) wgt[72] = bf[0];

    const float* mb = meanp + (long)b * HW_;
    const float* xb = maxp  + (long)b * HW_;
    for (int i = tid; i < 400; i += 256) {
        int ty = i / 20, tx = i % 20;
        int gy = by + ty - 2, gx = bx + tx - 2;
        bool in = (gy >= 0) & (gy < H_) & (gx >= 0) & (gx < W_);
        tmn[ty][tx] = in ? mb[gy * W_ + gx] : 0.0f;
        tmx[ty][tx] = in ? xb[gy * W_ + gx] : 0.0f;
    }
    __syncthreads();

    // 3x3 conv (pad 1): input tap = (ly+2)+(dy-1) = ly+1+dy
    float s3 = wgt[70];
    #pragma unroll
    for (int dy = 0; dy < 3; ++dy)
        #pragma unroll
        for (int dx = 0; dx < 3; ++dx) {
            s3 += wgt[dy * 3 + dx]     * tmn[ly + 1 + dy][lx + 1 + dx];
            s3 += wgt[9 + dy * 3 + dx] * tmx[ly + 1 + dy][lx + 1 + dx];
        }
    float a3v = sigmoidf_(s3);

    // 5x5 conv (pad 2): input tap = (ly+2)+(dy-2) = ly+dy
    float s5 = wgt[71];
    #pragma unroll
    for (int dy = 0; dy < 5; ++dy)
        #pragma unroll
        for (int dx = 0; dx < 5; ++dx) {
            s5 += wgt[18 + dy * 5 + dx]      * tmn[ly + dy][lx + dx];
            s5 += wgt[18 + 25 + dy * 5 + dx] * tmx[ly + dy][lx + dx];
        }
    float a5v = sigmoidf_(s5);

    float fv    = wgt[68] * a3v + wgt[69] * a5v + wgt[72];
    float scale = 0.2f + 0.8f * fv;   // MIN_SCALE + (MAX-MIN)*fused

    long p = (long)b * HW_ + (by + ly) * W_ + (bx + lx);
    fused[p] = scale;

    red[tid] = scale * dotp[p];
    __syncthreads();
    #pragma unroll
    for (int off = 128; off > 0; off >>= 1) {
        if (tid < off) red[tid] += red[tid + off];
        __syncthreads();
    }
    if (tid == 0)
        __hip_atomic_fetch_add(&pooledAcc[b], red[0],
                               __ATOMIC_RELAXED, __HIP_MEMORY_SCOPE_AGENT);
}

// ---------------------------------------------------------------------------
// k3: per-batch channel-attention MLP, batched across B=16 as a matrix product
//     CW[64,16] = sigmoid( Wc2[64,32] @ H[32,16] + bc2 )
//     computed with V_WMMA_F32_16X16X4_F32 (4 M-tiles x 8 K-steps), one wave32.
// ---------------------------------------------------------------------------
__global__ void k3_mlp(const float* __restrict__ pooledAcc, const float* __restrict__ br,
                       const float* __restrict__ wc1, const float* __restrict__ bc1,
                       const float* __restrict__ wc2, const float* __restrict__ bc2,
                       float* __restrict__ cw) {
    __shared__ float pooled[B_];
    __shared__ float Hm[HID_][B_ + 1];
    int lane = threadIdx.x;   // 32 threads = 1 wave

    if (lane < B_)
        pooled[lane] = pooledAcc[lane] * (1.0f / (float)HW_) + br[0];
    __syncthreads();

    {   // h[j,b] = relu(wc1[j]*pooled[b] + bc1[j])
        float w = wc1[lane], bb = bc1[lane];
        for (int b = 0; b < B_; ++b)
            Hm[lane][b] = fmaxf(w * pooled[b] + bb, 0.0f);
    }
    __syncthreads();

    // WMMA fragment lane mapping (wave32, 16x16x4 f32):
    //  A 16x4:  VGPR0 = K{0|2}, VGPR1 = K{1|3}; lanes 0-15 K+0/1, lanes 16-31 K+2/3, M=lane%16
    //  B 4x16:  same K striping; N = lane%16
    //  C 16x16: VGPR r -> M = r + 8*(lane/16), N = lane%16
    int half = lane >> 4;
    int l    = lane & 15;

    for (int m = 0; m < 4; ++m) {
        int mbase = m * 16;
        v8f acc = {0.f, 0.f, 0.f, 0.f, 0.f, 0.f, 0.f, 0.f};
        #pragma unroll
        for (int k0 = 0; k0 < 8; ++k0) {
            int kb = k0 * 4 + 2 * half;
            v2f a, bfrag;
            a.x     = wc2[(mbase + l) * HID_ + kb];
            a.y     = wc2[(mbase + l) * HID_ + kb + 1];
            bfrag.x = Hm[kb][l];
            bfrag.y = Hm[kb + 1][l];
            acc = __builtin_amdgcn_wmma_f32_16x16x4_f32(
                false, a, false, bfrag, (short)0, acc, false, false);
        }
        #pragma unroll
        for (int r = 0; r < 8; ++r) {
            int row = mbase + r + 8 * half;          // output channel c
            float v = sigmoidf_(acc[r] + bc2[row]);
            cw[l * C_ + row] = v;                    // cw[b=N, c]
        }
    }
}

// ---------------------------------------------------------------------------
// k4: out = x * (fused * cw[b,c] + 0.7), float4 streaming pass.
// ---------------------------------------------------------------------------
__global__ void k4_out(const float* __restrict__ x, const float* __restrict__ fused,
                       const float* __restrict__ cw, float* __restrict__ out) {
    long q = (long)blockIdx.x * blockDim.x + threadIdx.x;   // float4 index
    long p = q * 4;
    int  b   = (int)(p / ((long)C_ * HW_));
    long rem = p - (long)b * C_ * HW_;
    int  c   = (int)(rem / HW_);
    int  hw  = (int)(rem % HW_);

    const long total = (long)B_ * C_ * HW_;
    if (p + 8192 < total) __builtin_prefetch(x + p + 8192, 0, 1);

    float  s  = cw[b * C_ + c];                       // block-uniform -> scalar load
    float4 f  = *(const float4*)(fused + (long)b * HW_ + hw);
    float4 xv = *(const float4*)(x + p);
    float4 o;
    o.x = xv.x * (f.x * s + 0.7f);
    o.y = xv.y * (f.y * s + 0.7f);
    o.z = xv.z * (f.z * s + 0.7f);
    o.w = xv.w * (f.w * s + 0.7f);
    *(float4*)(out + p) = o;
}

// ---------------------------------------------------------------------------
extern "C" void kernel_launch(void* const* d_in, const int* in_sizes, int n_in,
                              void* d_out, int out_size, void* d_ws, size_t ws_size,
                              hipStream_t stream) {
    const float* x   = (const float*)d_in[0];
    const float* w3  = (const float*)d_in[1];
    const float* b3  = (const float*)d_in[2];
    const float* w5  = (const float*)d_in[3];
    const float* b5  = (const float*)d_in[4];
    const float* wf  = (const float*)d_in[5];
    const float* bfp = (const float*)d_in[6];
    const float* wr  = (const float*)d_in[7];
    const float* br  = (const float*)d_in[8];
    const float* wc1 = (const float*)d_in[9];
    const float* bc1 = (const float*)d_in[10];
    const float* wc2 = (const float*)d_in[11];
    const float* bc2 = (const float*)d_in[12];
    float* out = (float*)d_out;

    const long NPIX = (long)B_ * HW_;        // 1,048,576
    float* ws        = (float*)d_ws;
    float* meanp     = ws;
    float* maxp      = ws + NPIX;
    float* dotp      = ws + 2 * NPIX;
    float* fusedp    = ws + 3 * NPIX;
    float* pooledAcc = ws + 4 * NPIX;        // 16 floats
    float* cw        = ws + 4 * NPIX + 32;   // 16*64 floats

    k0_init<<<1, 32, 0, stream>>>(pooledAcc);

    k1_stats<<<(int)(NPIX / 4 / 256), 256, 0, stream>>>(x, wr, meanp, maxp, dotp);

    dim3 g2(W_ / 16, H_ / 16, B_), t2(16, 16);
    k2_conv<<<g2, t2, 0, stream>>>(meanp, maxp, dotp, w3, b3, w5, b5, wf, bfp,
                                   fusedp, pooledAcc);

    k3_mlp<<<1, 32, 0, stream>>>(pooledAcc, br, wc1, bc1, wc2, bc2, cw);

    const long NELEM = (long)B_ * C_ * HW_;  // 67,108,864
    k4_out<<<(int)(NELEM / 4 / 256), 256, 0, stream>>>(x, fusedp, cw, out);
}